// DecoderSourceTarget_33268816675215
// MI455X (gfx1250) — compile-verified
//
#include <hip/hip_runtime.h>
#include <stdint.h>

// ---------------------------------------------------------------------------
// DecoderSourceTarget: per-edge dot( x[i0, :128], x[i1, 128:] ) -> sigmoid
//
// MI455X / gfx1250 plan:
//  * 1 wave32 = 16 edges. Gather 16 src + 16 dst rows (512 B each, fully
//    coalesced: 32 lanes x b128) into LDS with GLOBAL_LOAD_ASYNC_TO_LDS_B128
//    in GVS form: scalar row base in SGPRs, loop-invariant VGPR operands,
//    per-row LDS stride carried in the instruction offset (which adds to both
//    the LDS and global address, so the global base pre-subtracts it).
//  * C = S(16x128) x D^T(128x16) via 32x V_WMMA_F32_16X16X4_F32 (fp32-exact),
//    two accumulators for ILP; diagonal of C = the 16 edge logits.
//  * Spill C through LDS, pull diagonal, sigmoid (v_exp_f32), coalesced store.
// Memory-bound: ~1.05 GB gathered reads/call; x (102 MB) is L2-resident
// (192 MB L2). This revision removes the per-row v_add_nc_u64 VALU address
// math that was co-limiting instruction issue.
// ---------------------------------------------------------------------------

typedef float v2f __attribute__((ext_vector_type(2)));
typedef float v8f __attribute__((ext_vector_type(8)));

#define HIDDEN 256
#define HALF   128
#define TILE   16
#define WAVES_PER_BLOCK 2
#define THREADS (WAVES_PER_BLOCK * 32)
// per-wave LDS: 16 src rows + 16 dst rows of 128 f32 = 16 KB
#define WAVE_LDS_F (TILE * HALF * 2)

// One gathered row pair per step. R is constexpr so offset:imm is legal.
// Global addr = SADDR + VADDR + offset ; LDS addr = VDST + offset.
// We want LDS += R*512 per row, so the scalar base subtracts R*512.
template <int R>
__device__ __forceinline__ void gather_rows(int vidx, unsigned vls, unsigned vld,
                                            unsigned voff, uint64_t xb)
{
    if constexpr (R < TILE) {
        int si = __builtin_amdgcn_readlane(vidx, R);        // -> SGPR
        int di = __builtin_amdgcn_readlane(vidx, R + 16);   // -> SGPR
        uint64_t bs = xb + ((uint64_t)(unsigned)si << 10)
                         - (uint64_t)(R * 512);                       // src cols 0..127
        uint64_t bd = xb + ((uint64_t)(unsigned)di << 10) + (HALF * 4)
                         - (uint64_t)(R * 512);                       // dst cols 128..255
        asm volatile("global_load_async_to_lds_b128 %0, %1, %2 offset:%3"
                     :: "v"(vls), "v"(voff), "s"(bs), "n"(R * 512) : "memory");
        asm volatile("global_load_async_to_lds_b128 %0, %1, %2 offset:%3"
                     :: "v"(vld), "v"(voff), "s"(bd), "n"(R * 512) : "memory");
        gather_rows<R + 1>(vidx, vls, vld, voff, xb);
    }
}

__global__ __launch_bounds__(THREADS)
void decoder_dot_wmma(const float* __restrict__ x,
                      const long long* __restrict__ eidx,
                      float* __restrict__ out,
                      int n_edges)
{
    __shared__ float smem[WAVES_PER_BLOCK * WAVE_LDS_F];

    const int lane = threadIdx.x & 31;
    const int wave = threadIdx.x >> 5;
    const int tile = blockIdx.x * WAVES_PER_BLOCK + wave;
    const int ebase = tile * TILE;
    if (ebase >= n_edges) return;            // wave-uniform guard (EXEC stays all-1s)

    // lanes 0-15: src index of edge ebase+lane ; lanes 16-31: dst index (int64 fits i32)
    long long ei = (lane < 16) ? eidx[ebase + lane]
                               : eidx[n_edges + ebase + (lane - 16)];
    int vidx = (int)ei;

    float* wbase = &smem[wave * WAVE_LDS_F];
    // low 32 bits of a flat shared pointer == wave-relative LDS byte offset
    const unsigned ldsSrc = (unsigned)(uintptr_t)wbase;
    const unsigned ldsDst = ldsSrc + TILE * HALF * 4;
    const uint64_t xb  = (uint64_t)(uintptr_t)x;
    const unsigned voff = (unsigned)(lane * 16);     // VADDR, loop-invariant
    const unsigned vls  = ldsSrc + voff;             // VDST src region, loop-invariant
    const unsigned vld  = ldsDst + voff;             // VDST dst region, loop-invariant

    gather_rows<0>(vidx, vls, vld, voff, xb);
    asm volatile("s_wait_asynccnt 0" ::: "memory");

    const float* srcS = wbase;                 // [16][128] src tile (A, row-major)
    const float* dstS = wbase + TILE * HALF;   // [16][128] dst tile (B = D^T chunks)

#if __has_builtin(__builtin_amdgcn_wmma_f32_16x16x4_f32)
    // A 16x4 f32 layout (ISA 7.12.2): lanes 0-15 hold K=0,1 ; lanes 16-31 hold K=2,3.
    // B 4x16 mirrors the same per-lane K mapping (lane half -> K pair).
    const int row  = lane & 15;
    const int koff = (lane >> 4) << 1;
    const float* sA = srcS + row * HALF + koff;
    const float* sB = dstS + row * HALF + koff;

    v8f acc0 = {0.f, 0.f, 0.f, 0.f, 0.f, 0.f, 0.f, 0.f};
    v8f acc1 = acc0;
#pragma unroll
    for (int k = 0; k < HALF / 4; k += 2) {
        v2f a0 = *(const v2f*)(sA + 4 * k);
        v2f b0 = *(const v2f*)(sB + 4 * k);
        v2f a1 = *(const v2f*)(sA + 4 * (k + 1));
        v2f b1 = *(const v2f*)(sB + 4 * (k + 1));
        acc0 = __builtin_amdgcn_wmma_f32_16x16x4_f32(false, a0, false, b0,
                                                     (short)0, acc0, false, false);
        acc1 = __builtin_amdgcn_wmma_f32_16x16x4_f32(false, a1, false, b1,
                                                     (short)0, acc1, false, false);
    }
    v8f acc = acc0 + acc1;

    // spill the 16x16 C tile and pull its diagonal:
    // C[m][n]: vgpr m%8, lane n + 16*(m/8)  ->  stored at wbase[lane*8 + vgpr]
#pragma unroll
    for (int v = 0; v < 8; ++v) wbase[lane * 8 + v] = acc[v];

    if (lane < 16) {
        int m = lane;
        float dot = (m < 8) ? wbase[m * 8 + m]
                            : wbase[(m + 16) * 8 + (m - 8)];
        out[ebase + m] = 1.0f / (1.0f + __expf(-dot));
    }
#else
    // scalar fallback (should not trigger on gfx1250): lane m reduces edge m
    if (lane < 16) {
        float dot = 0.f;
        for (int k = 0; k < HALF; ++k)
            dot += srcS[lane * HALF + k] * dstS[lane * HALF + k];
        out[ebase + lane] = 1.0f / (1.0f + __expf(-dot));
    }
#endif
}

extern "C" void kernel_launch(void* const* d_in, const int* in_sizes, int n_in,
                              void* d_out, int out_size, void* d_ws, size_t ws_size,
                              hipStream_t stream) {
    const float*     x    = (const float*)d_in[0];
    const long long* eidx = (const long long*)d_in[1];   // int64 (2, n_edges)
    float*           out  = (float*)d_out;

    const int n_edges = in_sizes[1] / 2;                 // 1,000,000
    const int ntiles  = (n_edges + TILE - 1) / TILE;     // 62,500
    const int blocks  = (ntiles + WAVES_PER_BLOCK - 1) / WAVES_PER_BLOCK;

    decoder_dot_wmma<<<blocks, THREADS, 0, stream>>>(x, eidx, out, n_edges);
}